// TsGen_47553877901968
// MI455X (gfx1250) — compile-verified
//
#include <hip/hip_runtime.h>
#include <hip/hip_bf16.h>

#define DEVFN __device__ __forceinline__

typedef __attribute__((ext_vector_type(2))) float v2f;
typedef __attribute__((ext_vector_type(8))) float v8f;

constexpr int B = 8, N = 192, BN = B * N;
constexpr int ATOM_F = 16, BOND_F = 5;
constexpr int XS = 128, XV = 16, ES = 64, EV = 8;
constexpr int H = 8, DH = XS / H;
constexpr long NP = (long)B * N * N;      // 294912 pair rows
constexpr float CLAMP_V = 10.0f;

DEVFN float sigm_(float x) { return 1.0f / (1.0f + expf(-x)); }
DEVFN float silu_(float x) { return x * sigm_(x); }

// ---------------- CDNA5 async global->LDS staging ----------------------------
#if defined(__has_builtin)
#if __has_builtin(__builtin_amdgcn_global_load_async_to_lds_b32) && \
    __has_builtin(__builtin_amdgcn_s_wait_asynccnt)
#define USE_ASYNC_LDS 1
#endif
#endif

typedef __attribute__((address_space(1))) int g_i32;
typedef __attribute__((address_space(3))) int l_i32;

DEVFN void async_ld32(float* l, const float* g) {
#ifdef USE_ASYNC_LDS
  __builtin_amdgcn_global_load_async_to_lds_b32((g_i32*)g, (l_i32*)l, 0, 0);
#else
  *l = *g;
#endif
}
DEVFN void async_fence() {
#ifdef USE_ASYNC_LDS
  __builtin_amdgcn_s_wait_asynccnt(0);
#endif
}

// ---------------- CDNA5 WMMA fp32 16x16x4 ------------------------------------
// A (16x4): lane&15 = M row, lane>>4 selects K pair {0,1}/{2,3}
// B (4x16): lane&15 = N col, lane>>4 selects K pair
// C/D (16x16): VGPR r holds row r (lanes 0-15) / row r+8 (lanes 16-31), col=lane&15
DEVFN v8f wmma4(v2f a, v2f b, v8f c) {
  return __builtin_amdgcn_wmma_f32_16x16x4_f32(false, a, false, b, (short)0, c,
                                               false, false);
}

// ---------------- A-matrix loaders -------------------------------------------
struct DirectA {
  const float* A; int lda; long sz;
  DEVFN float ld(int z, long m, int k) const {
    return A[(long)z * sz + m * (long)lda + k];
  }
};
struct ConcatA {
  const float* A1; const float* A2; int lda1, lda2, K1;
  DEVFN float ld(int /*z*/, long m, int k) const {
    return (k < K1) ? A1[m * (long)lda1 + k] : A2[m * (long)lda2 + (k - K1)];
  }
};

// ---------------- generic WMMA GEMM ------------------------------------------
// 4 waves / block, 64 rows / block, one 16-col N tile / block.
// The K x 16 weight panel is staged once into LDS via async global->LDS copies
// and shared by all 4 waves.
// C[z][m][n] (acc?+=:=) act( sum_k A(z,m,k)*W[z][k][n] + bias[n] ) * rowScale[m]
template <typename AL>
__global__ void gemm_wmma_kernel(AL a, const float* __restrict__ W, int ldb,
                                 long sW, const float* __restrict__ bias,
                                 float* __restrict__ C, int ldc, long sC,
                                 long M, int Nc, int K, int act, int acc,
                                 const float* __restrict__ rowScale) {
  __shared__ float Wl[4096];  // K4 x 16 panel, K <= 256
  const int z = blockIdx.z;
  const int nBase = blockIdx.y * 16;
  const int tid = threadIdx.x;  // 0..127
  const float* Wz = W + (long)z * sW;
  const int K4 = (K + 3) & ~3;
  const int tot = K4 * 16;
  for (int idx = tid; idx < tot; idx += 128) {
    const int kk = idx >> 4;
    const int n = nBase + (idx & 15);
    if (kk < K && n < Nc)
      async_ld32(&Wl[idx], &Wz[(long)kk * ldb + n]);
    else
      Wl[idx] = 0.f;
  }
  async_fence();
  __syncthreads();

  const int wave = tid >> 5;
  const int lane = tid & 31;
  const int half = lane >> 4;
  const int ml = lane & 15;
  const long mBase = ((long)blockIdx.x * 4 + wave) * 16;
  const long m = mBase + ml;
  const bool mok = (m < M);
  const int n = nBase + ml;
  const bool nok = (n < Nc);
  v8f c = {0.f, 0.f, 0.f, 0.f, 0.f, 0.f, 0.f, 0.f};
  for (int k0 = 0; k0 < K; k0 += 4) {
    const int ka = k0 + half * 2;
    v2f av, bv;
    av.x = (mok && ka < K) ? a.ld(z, m, ka) : 0.f;
    av.y = (mok && (ka + 1) < K) ? a.ld(z, m, ka + 1) : 0.f;
    bv.x = Wl[ka * 16 + ml];
    bv.y = Wl[(ka + 1) * 16 + ml];
    c = wmma4(av, bv, c);
  }
  if (nok) {
    const float bn = bias ? bias[n] : 0.f;
    float* Cz = C + (long)z * sC;
#pragma unroll
    for (int r = 0; r < 8; ++r) {
      long row = mBase + r + 8 * half;
      if (row >= M) continue;
      float vv = c[r] + bn;
      if (act == 1) vv = silu_(vv);
      else if (act == 2) vv = sigm_(vv);
      if (rowScale) vv *= rowScale[row];
      long o = row * (long)ldc + n;
      if (acc) Cz[o] += vv; else Cz[o] = vv;
    }
  }
}

// ---------------- fused pair e_sca update ------------------------------------
// e += ( silu(e + si_i + sj_j) @ We + be ) * pair_m, in place.
// 4 waves / block; each wave owns 16 consecutive pair rows and ALL 4 column
// tiles (ES=64), so the in-place update has no cross-wave hazard. We (64x64)
// + be are staged in LDS (stride 80 -> the two EXEC halves hit disjoint bank
// sets) via async global->LDS copies.
constexpr int WE_LD = 80;
__global__ void pair_e_update_kernel(float* __restrict__ e,
                                     const float* __restrict__ si,
                                     const float* __restrict__ sj,
                                     const float* __restrict__ We,
                                     const float* __restrict__ be,
                                     const float* __restrict__ pairm) {
  __shared__ float WeL[ES * WE_LD];
  __shared__ float beL[ES];
  const int tid = threadIdx.x;  // 0..127
  for (int idx = tid; idx < ES * ES; idx += 128) {
    const int kk = idx >> 6, cc2 = idx & 63;
    async_ld32(&WeL[kk * WE_LD + cc2], &We[idx]);
  }
  if (tid < ES) async_ld32(&beL[tid], &be[tid]);
  async_fence();
  __syncthreads();

  const int wave = tid >> 5;
  const int lane = tid & 31;
  const int half = lane >> 4;
  const int ml = lane & 15;
  const long mBase = ((long)blockIdx.x * 4 + wave) * 16;
  const long m = mBase + ml;
  const int j = (int)(m % N);
  const long bi = m / N;       // b*N + i
  const long bb = bi / N;      // b
  const float* er = e + m * ES;
  const float* sr = si + bi * ES;
  const float* tr = sj + (bb * N + j) * ES;
  v8f z8 = {0.f, 0.f, 0.f, 0.f, 0.f, 0.f, 0.f, 0.f};
  v8f cc[4] = {z8, z8, z8, z8};
  for (int k0 = 0; k0 < ES; k0 += 4) {
    const int ka = k0 + half * 2;
    v2f av;
    av.x = silu_(er[ka] + sr[ka] + tr[ka]);
    av.y = silu_(er[ka + 1] + sr[ka + 1] + tr[ka + 1]);
#pragma unroll
    for (int t = 0; t < 4; ++t) {
      v2f bv;
      bv.x = WeL[ka * WE_LD + t * 16 + ml];
      bv.y = WeL[(ka + 1) * WE_LD + t * 16 + ml];
      cc[t] = wmma4(av, bv, cc[t]);
    }
  }
#pragma unroll
  for (int t = 0; t < 4; ++t) {
    const int n = t * 16 + ml;
    const float bn = beL[n];
#pragma unroll
    for (int r = 0; r < 8; ++r) {
      long row = mBase + r + 8 * half;
      long o = row * ES + n;
      e[o] += (cc[t][r] + bn) * pairm[row];
    }
  }
}

// ---------------- attention: fused logits + masked softmax -------------------
__global__ void attn_kernel(const float* __restrict__ q,
                            const float* __restrict__ k,
                            const float* __restrict__ biasp,
                            const float* __restrict__ mf,
                            float* __restrict__ attn) {
  const int i = blockIdx.x, hh = blockIdx.y, b = blockIdx.z;
  const int j = threadIdx.x;
  __shared__ float qs[DH];
  __shared__ float red[N];
  if (j < DH) qs[j] = q[((long)(b * N + i) * H + hh) * DH + j];
  __syncthreads();
  const float* kr = k + ((long)(b * N + j) * H + hh) * DH;
  float l = 0.f;
#pragma unroll
  for (int d = 0; d < DH; ++d) l += qs[d] * kr[d];
  l *= 0.25f;  // 1/sqrt(DH)
  l += biasp[((long)(b * N + i) * N + j) * H + hh];
  if (mf[b * N + j] == 0.f) l = -1.0e9f;
  red[j] = l;
  __syncthreads();
  for (int s = 128; s > 0; s >>= 1) {
    if (j < s && j + s < N) red[j] = fmaxf(red[j], red[j + s]);
    __syncthreads();
  }
  const float mx = red[0];
  __syncthreads();
  const float ev = expf(l - mx);
  red[j] = ev;
  __syncthreads();
  for (int s = 128; s > 0; s >>= 1) {
    if (j < s && j + s < N) red[j] += red[j + s];
    __syncthreads();
  }
  attn[((long)(b * H + hh) * N + i) * N + j] = ev / red[0];
}

// ---------------- small elementwise kernels ----------------------------------
__global__ void mask_init_kernel(const unsigned char* mk, float* mf) {
  long i = blockIdx.x * 256L + threadIdx.x;
  if (i < BN) mf[i] = mk[i] ? 1.0f : 0.0f;
}
__global__ void msum_kernel(const float* mf, float* msum) {
  int b = threadIdx.x;
  if (b >= B) return;
  float s = 0.f;
  for (int n2 = 0; n2 < N; ++n2) s += mf[b * N + n2];
  msum[b] = s;
}
__global__ void pairm_kernel(const float* mf, float* pm) {
  long idx = blockIdx.x * 256L + threadIdx.x;
  if (idx >= NP) return;
  int j = (int)(idx % N);
  long t = idx / N;
  int i = (int)(t % N);
  int b = (int)(t / N);
  pm[idx] = mf[b * N + i] * mf[b * N + j];
}
__global__ void cmean_kernel(const float* coords, float* cmean) {
  int t = threadIdx.x;
  if (t >= B * 3) return;
  int b = t / 3, d = t % 3;
  float s = 0.f;
  for (int n2 = 0; n2 < N; ++n2) s += coords[((long)b * N + n2) * 3 + d];
  cmean[t] = s / (float)N;
}
__global__ void xvec_init_kernel(const float* coords, const float* cmean,
                                 const float* Wxv, float* xv) {
  long idx = blockIdx.x * 256L + threadIdx.x;
  if (idx >= (long)BN * XV * 3) return;
  int d = (int)(idx % 3);
  long t = idx / 3;
  int o = (int)(t % XV);
  long r = t / XV;
  int b = (int)(r / N);
  float cc = coords[r * 3 + d] - cmean[b * 3 + d];
  xv[idx] = cc * Wxv[o];
}
__global__ void ln_kernel(const float* __restrict__ x, float* __restrict__ y) {
  __shared__ float red[XS];
  long r = blockIdx.x;
  int t = threadIdx.x;
  float v = x[r * XS + t];
  red[t] = v;
  __syncthreads();
  for (int s = 64; s > 0; s >>= 1) { if (t < s) red[t] += red[t + s]; __syncthreads(); }
  float mu = red[0] / (float)XS;
  __syncthreads();
  float d = v - mu;
  red[t] = d * d;
  __syncthreads();
  for (int s = 64; s > 0; s >>= 1) { if (t < s) red[t] += red[t + s]; __syncthreads(); }
  float var = red[0] / (float)XS;
  y[r * XS + t] = d * rsqrtf(var + 1e-5f);
}
__global__ void attnm_kernel(const float* attn, float* attnm) {
  long idx = blockIdx.x * 256L + threadIdx.x;
  if (idx >= NP) return;
  int j = (int)(idx % N);
  long t = idx / N;
  int i = (int)(t % N);
  int b = (int)(t / N);
  float s = 0.f;
  for (int hh = 0; hh < H; ++hh)
    s += attn[((long)(b * H + hh) * N + i) * N + j];
  attnm[idx] = s * (1.0f / H);
}
// v (b,j,h,d) -> vt (b,h,j,d)
__global__ void trans_v_kernel(const float* v, float* vt) {
  long idx = blockIdx.x * 256L + threadIdx.x;
  if (idx >= (long)BN * XS) return;
  int d = (int)(idx % DH);
  long r = idx / DH;
  int hh = (int)(r % H);
  long r2 = r / H;
  int j = (int)(r2 % N);
  int b = (int)(r2 / N);
  vt[(((long)b * H + hh) * N + j) * DH + d] = v[idx];
}
// satt_t (b,h,i,d) -> scaatt (b,i,h,d)
__global__ void trans_att_kernel(const float* st, float* sc) {
  long idx = blockIdx.x * 256L + threadIdx.x;
  if (idx >= (long)BN * XS) return;
  int d = (int)(idx % DH);
  long r = idx / DH;
  int i = (int)(r % N);
  long r2 = r / N;
  int hh = (int)(r2 % H);
  int b = (int)(r2 / H);
  sc[(((long)b * N + i) * H + hh) * DH + d] = st[idx];
}
__global__ void vecmix_kernel(const float* xv, const float* W, float* out) {
  long idx = blockIdx.x * 256L + threadIdx.x;
  if (idx >= (long)BN * XV * 3) return;
  int d = (int)(idx % 3);
  long t = idx / 3;
  int o = (int)(t % XV);
  long row = t / XV;
  float s = 0.f;
  for (int i2 = 0; i2 < XV; ++i2)
    s += xv[(row * XV + i2) * 3 + d] * W[i2 * XV + o];
  out[idx] = s;
}
__global__ void vnorm_kernel(const float* xv, float* vn) {
  long idx = blockIdx.x * 256L + threadIdx.x;
  if (idx >= (long)BN * XV) return;
  float a = xv[idx * 3 + 0], b = xv[idx * 3 + 1], c = xv[idx * 3 + 2];
  vn[idx] = sqrtf(a * a + b * b + c * c + 1e-8f);
}
__global__ void addsca_kernel(float* xs, const float* ffo, const float* mf) {
  long idx = blockIdx.x * 256L + threadIdx.x;
  if (idx >= (long)BN * XS) return;
  long row = idx / XS;
  int c = (int)(idx % XS);
  xs[idx] += ffo[row * (XS + XV) + c] * mf[row];
}
__global__ void vecgate_kernel(float* xv, const float* ffo) {
  long idx = blockIdx.x * 256L + threadIdx.x;
  if (idx >= (long)BN * XV * 3) return;
  long t = idx / 3;
  int c = (int)(t % XV);
  long row = t / XV;
  xv[idx] *= sigm_(ffo[row * (XS + XV) + XS + c]);
}
__global__ void wc_kernel(const float* es, const float* Wc, const float* bc,
                          const float* pairm, float* wbuf) {
  long idx = blockIdx.x * 256L + threadIdx.x;
  if (idx >= NP) return;
  const float* er = es + idx * ES;
  __builtin_prefetch(er + 64 * ES, 0, 0);  // global_prefetch_b8, stream ahead
  float s = bc[0];
  for (int kk = 0; kk < ES; ++kk) s += er[kk] * Wc[kk];
  wbuf[idx] = s * pairm[idx];
}
__global__ void delta_kernel(const float* wb, const float* coords,
                             const float* msum, float* dbuf) {
  const int i = blockIdx.x, b = blockIdx.y, j = threadIdx.x;
  __shared__ float sx[N], sy[N], sz[N];
  long row = (long)b * N + i;
  float w = wb[row * N + j];
  const float* ci = coords + row * 3;
  const float* cj = coords + ((long)b * N + j) * 3;
  sx[j] = w * (ci[0] - cj[0]);
  sy[j] = w * (ci[1] - cj[1]);
  sz[j] = w * (ci[2] - cj[2]);
  __syncthreads();
  for (int s = 128; s > 0; s >>= 1) {
    if (j < s && j + s < N) { sx[j] += sx[j + s]; sy[j] += sy[j + s]; sz[j] += sz[j + s]; }
    __syncthreads();
  }
  if (j == 0) {
    float dn = msum[b] + 1e-6f;
    dbuf[row * 3 + 0] = fminf(fmaxf(sx[0] / dn, -CLAMP_V), CLAMP_V);
    dbuf[row * 3 + 1] = fminf(fmaxf(sy[0] / dn, -CLAMP_V), CLAMP_V);
    dbuf[row * 3 + 2] = fminf(fmaxf(sz[0] / dn, -CLAMP_V), CLAMP_V);
  }
}
__global__ void coords_add_kernel(float* coords, const float* dbuf,
                                  const float* mf) {
  long idx = blockIdx.x * 256L + threadIdx.x;
  if (idx >= (long)BN * 3) return;
  coords[idx] += dbuf[idx] * mf[idx / 3];
}
__global__ void xgate_sca_kernel(const float* g, float* xr, const float* xp) {
  long idx = blockIdx.x * 256L + threadIdx.x;
  if (idx >= (long)BN * XS) return;
  long row = idx / XS;
  int c = (int)(idx % XS);
  float gg = g[row * (XS + XV) + c];
  xr[idx] = gg * xr[idx] + (1.f - gg) * xp[idx];
}
__global__ void xgate_vec_kernel(const float* g, float* vr, const float* vp) {
  long idx = blockIdx.x * 256L + threadIdx.x;
  if (idx >= (long)BN * XV * 3) return;
  long t = idx / 3;
  int c = (int)(t % XV);
  long row = t / XV;
  float gg = g[row * (XS + XV) + XS + c];
  vr[idx] = gg * vr[idx] + (1.f - gg) * vp[idx];
}
__global__ void egate_combine_kernel(const float* g, float* er, const float* ep,
                                     long base, long count) {
  long idx = blockIdx.x * 256L + threadIdx.x;
  if (idx >= count) return;
  float gg = g[idx];
  long o = base + idx;
  er[o] = gg * er[o] + (1.f - gg) * ep[o];
}
__global__ void sigemb_kernel(float* xs, const float* sig, const float* W,
                              const float* bvec) {
  long idx = blockIdx.x * 256L + threadIdx.x;
  if (idx >= (long)BN * XS) return;
  long row = idx / XS;
  int c = (int)(idx % XS);
  int b = (int)(row / N);
  xs[idx] += sig[b] * W[c] + bvec[c];
}
__global__ void copy_kernel(float* dst, const float* src, long n) {
  long idx = blockIdx.x * 256L + threadIdx.x;
  if (idx < n) dst[idx] = src[idx];
}
__global__ void final_kernel(const float* ts, const float* coords,
                             const float* sig, float* out) {
  long idx = blockIdx.x * 256L + threadIdx.x;
  if (idx >= (long)BN * 3) return;
  int b = (int)((idx / 3) / N);
  float s = sig[b];
  float cs = 0.25f / ((s - 0.002f) * (s - 0.002f) + 0.25f);
  float co = 0.5f * (s - 0.002f) * rsqrtf(s * s + 0.25f);
  out[idx] = cs * ts[idx] + co * coords[idx];
}

// ---------------- host side --------------------------------------------------
struct Lin { const float* W; const float* b; };
struct LayerP {
  Lin W1, W2, Wb, Wc, We, Wei, Wej; const float* Wev;
  Lin Wk, Wo, Wq, Wv; const float* Wvv;
};
struct EncP { Lin e_sca; const float* e_vec; LayerP lay[2]; Lin x_sca; const float* x_vec; };

static int parse_lin(void* const* din, int i, Lin& L) {
  L.W = (const float*)din[i]; L.b = (const float*)din[i + 1]; return i + 2;
}
static int parse_layer(void* const* din, int i, LayerP& L) {
  // jax tree_flatten (sorted keys): W1 W2 Wb Wc We Wei Wej Wev Wk Wo Wq Wv Wvv
  i = parse_lin(din, i, L.W1); i = parse_lin(din, i, L.W2);
  i = parse_lin(din, i, L.Wb); i = parse_lin(din, i, L.Wc);
  i = parse_lin(din, i, L.We); i = parse_lin(din, i, L.Wei);
  i = parse_lin(din, i, L.Wej); L.Wev = (const float*)din[i++];
  i = parse_lin(din, i, L.Wk); i = parse_lin(din, i, L.Wo);
  i = parse_lin(din, i, L.Wq); i = parse_lin(din, i, L.Wv);
  L.Wvv = (const float*)din[i++];
  return i;
}
static int parse_enc(void* const* din, int i, EncP& E) {
  // sorted: e_sca e_vec layers x_sca x_vec
  i = parse_lin(din, i, E.e_sca); E.e_vec = (const float*)din[i++];
  i = parse_layer(din, i, E.lay[0]); i = parse_layer(din, i, E.lay[1]);
  i = parse_lin(din, i, E.x_sca); E.x_vec = (const float*)din[i++];
  return i;
}

struct WS {
  float *mf, *msum, *pairm, *cmean;
  float *xs_r, *xs_p, *xv_r, *xv_p, *es_r, *es_p;
  float *coords, *dbuf;
  float *h, *q, *k, *v, *vt, *satt_t, *scaatt;
  float *biasp, *attn, *attnm, *mixed, *vnorm, *ffh, *ffo, *si, *sj, *wbuf, *gbuf;
};

static inline unsigned gblk(long n) { return (unsigned)((n + 255) / 256); }
static inline dim3 gtile(long M, int Nc, int z) {
  return dim3((unsigned)((M + 63) / 64), (unsigned)((Nc + 15) / 16), (unsigned)z);
}

static void gemmD(hipStream_t st, const float* A, int lda, long sA,
                  const float* W, int ldb, long sW, const float* bias,
                  float* C, int ldc, long sC, long M, int Nc, int K, int batch,
                  int act, int acc, const float* rowScale) {
  gemm_wmma_kernel<<<gtile(M, Nc, batch), 128, 0, st>>>(
      DirectA{A, lda, sA}, W, ldb, sW, bias, C, ldc, sC, M, Nc, K, act, acc,
      rowScale);
}

static void run_layer(hipStream_t st, const LayerP& L, WS& w, float* xs,
                      float* xv, float* es, float* coords, bool doCoords) {
  // h = ln(xs); q,k,v
  ln_kernel<<<BN, XS, 0, st>>>(xs, w.h);
  gemmD(st, w.h, XS, 0, L.Wq.W, XS, 0, L.Wq.b, w.q, XS, 0, BN, XS, XS, 1, 0, 0, nullptr);
  gemmD(st, w.h, XS, 0, L.Wk.W, XS, 0, L.Wk.b, w.k, XS, 0, BN, XS, XS, 1, 0, 0, nullptr);
  gemmD(st, w.h, XS, 0, L.Wv.W, XS, 0, L.Wv.b, w.v, XS, 0, BN, XS, XS, 1, 0, 0, nullptr);
  // pair bias (b,i,j,h) = e_sca @ Wb
  gemmD(st, es, ES, 0, L.Wb.W, H, 0, L.Wb.b, w.biasp, H, 0, NP, H, ES, 1, 0, 0, nullptr);
  // attention softmax
  attn_kernel<<<dim3(N, H, B), N, 0, st>>>(w.q, w.k, w.biasp, w.mf, w.attn);
  // sca_att = attn @ v  (batched over b*h)
  trans_v_kernel<<<gblk((long)BN * XS), 256, 0, st>>>(w.v, w.vt);
  gemmD(st, w.attn, N, (long)N * N, w.vt, DH, (long)N * DH, nullptr, w.satt_t,
        DH, (long)N * DH, N, DH, N, B * H, 0, 0, nullptr);
  trans_att_kernel<<<gblk((long)BN * XS), 256, 0, st>>>(w.satt_t, w.scaatt);
  // xs += (sca_att@Wo + bo) * m
  gemmD(st, w.scaatt, XS, 0, L.Wo.W, XS, 0, L.Wo.b, xs, XS, 0, BN, XS, XS, 1,
        0, 1, w.mf);
  // xv += attn_mean @ vec_mix(xv, Wvv)
  attnm_kernel<<<gblk(NP), 256, 0, st>>>(w.attn, w.attnm);
  vecmix_kernel<<<gblk((long)BN * XV * 3), 256, 0, st>>>(xv, L.Wvv, w.mixed);
  gemmD(st, w.attnm, N, (long)N * N, w.mixed, XV * 3, (long)N * XV * 3,
        nullptr, xv, XV * 3, (long)N * XV * 3, N, XV * 3, N, B, 0, 1, nullptr);
  // FF
  vnorm_kernel<<<gblk((long)BN * XV), 256, 0, st>>>(xv, w.vnorm);
  ln_kernel<<<BN, XS, 0, st>>>(xs, w.h);
  gemm_wmma_kernel<<<gtile(BN, 2 * XS, 1), 128, 0, st>>>(
      ConcatA{w.h, w.vnorm, XS, XV, XS}, L.W1.W, 2 * XS, 0, L.W1.b, w.ffh,
      2 * XS, 0, (long)BN, 2 * XS, XS + XV, 1, 0, nullptr);
  gemmD(st, w.ffh, 2 * XS, 0, L.W2.W, XS + XV, 0, L.W2.b, w.ffo, XS + XV, 0,
        BN, XS + XV, 2 * XS, 1, 0, 0, nullptr);
  addsca_kernel<<<gblk((long)BN * XS), 256, 0, st>>>(xs, w.ffo, w.mf);
  vecgate_kernel<<<gblk((long)BN * XV * 3), 256, 0, st>>>(xv, w.ffo);
  // e_sca update (fused silu + WMMA, LDS-staged We)
  gemmD(st, xs, XS, 0, L.Wei.W, ES, 0, L.Wei.b, w.si, ES, 0, BN, ES, XS, 1, 0, 0, nullptr);
  gemmD(st, xs, XS, 0, L.Wej.W, ES, 0, L.Wej.b, w.sj, ES, 0, BN, ES, XS, 1, 0, 0, nullptr);
  pair_e_update_kernel<<<(unsigned)(NP / 64), 128, 0, st>>>(es, w.si, w.sj,
                                                            L.We.W, L.We.b,
                                                            w.pairm);
  if (doCoords) {  // encoder coords outputs are dead -> decoder only
    wc_kernel<<<gblk(NP), 256, 0, st>>>(es, L.Wc.W, L.Wc.b, w.pairm, w.wbuf);
    delta_kernel<<<dim3(N, B), N, 0, st>>>(w.wbuf, coords, w.msum, w.dbuf);
    coords_add_kernel<<<gblk((long)BN * 3), 256, 0, st>>>(coords, w.dbuf, w.mf);
  }
}

static void run_encoder(hipStream_t st, const EncP& E, WS& w,
                        const float* atoms, const float* adjs,
                        const float* coords_in, float* xs, float* xv,
                        float* es) {
  cmean_kernel<<<1, B * 3, 0, st>>>(coords_in, w.cmean);
  gemmD(st, atoms, ATOM_F, 0, E.x_sca.W, XS, 0, E.x_sca.b, xs, XS, 0, BN, XS,
        ATOM_F, 1, 0, 0, nullptr);
  xvec_init_kernel<<<gblk((long)BN * XV * 3), 256, 0, st>>>(coords_in, w.cmean,
                                                            E.x_vec, xv);
  gemmD(st, adjs, BOND_F, 0, E.e_sca.W, ES, 0, E.e_sca.b, es, ES, 0, NP, ES,
        BOND_F, 1, 0, 0, nullptr);
  for (int l = 0; l < 2; ++l)
    run_layer(st, E.lay[l], w, xs, xv, es, nullptr, false);
}

extern "C" void kernel_launch(void* const* d_in, const int* in_sizes, int n_in,
                              void* d_out, int out_size, void* d_ws,
                              size_t ws_size, hipStream_t stream) {
  (void)in_sizes; (void)n_in; (void)out_size; (void)ws_size;
  const float* ratoms = (const float*)d_in[0];
  const float* patoms = (const float*)d_in[1];
  const float* radjs = (const float*)d_in[2];
  const float* padjs = (const float*)d_in[3];
  const float* rcoords = (const float*)d_in[4];
  const float* pcoords = (const float*)d_in[5];
  const float* tscoords = (const float*)d_in[6];
  const float* sigmas = (const float*)d_in[7];
  const unsigned char* masks = (const unsigned char*)d_in[8];

  // params in jax tree_flatten order: dec_layers, e_gate, enc_p, enc_r,
  // sig_emb, x_gate
  int idx = 9;
  LayerP dec[2];
  idx = parse_layer(d_in, idx, dec[0]);
  idx = parse_layer(d_in, idx, dec[1]);
  Lin e_gate; idx = parse_lin(d_in, idx, e_gate);
  EncP enc_p; idx = parse_enc(d_in, idx, enc_p);
  EncP enc_r; idx = parse_enc(d_in, idx, enc_r);
  Lin sig_emb; idx = parse_lin(d_in, idx, sig_emb);
  Lin x_gate; idx = parse_lin(d_in, idx, x_gate);

  // workspace bump allocation (fp32)
  float* base = (float*)d_ws;
  size_t off = 0;
  auto alloc = [&](size_t n) { float* r = base + off; off += n; return r; };
  WS w;
  w.mf = alloc(BN); w.msum = alloc(B); w.pairm = alloc(NP); w.cmean = alloc(B * 3);
  w.xs_r = alloc((size_t)BN * XS); w.xs_p = alloc((size_t)BN * XS);
  w.xv_r = alloc((size_t)BN * XV * 3); w.xv_p = alloc((size_t)BN * XV * 3);
  w.es_r = alloc((size_t)NP * ES); w.es_p = alloc((size_t)NP * ES);
  w.coords = alloc((size_t)BN * 3); w.dbuf = alloc((size_t)BN * 3);
  w.h = alloc((size_t)BN * XS); w.q = alloc((size_t)BN * XS);
  w.k = alloc((size_t)BN * XS); w.v = alloc((size_t)BN * XS);
  w.vt = alloc((size_t)BN * XS); w.satt_t = alloc((size_t)BN * XS);
  w.scaatt = alloc((size_t)BN * XS);
  w.biasp = alloc((size_t)NP * H); w.attn = alloc((size_t)NP * H);
  w.attnm = alloc(NP);
  w.mixed = alloc((size_t)BN * XV * 3); w.vnorm = alloc((size_t)BN * XV);
  w.ffh = alloc((size_t)BN * 2 * XS); w.ffo = alloc((size_t)BN * (XS + XV));
  w.si = alloc((size_t)BN * ES); w.sj = alloc((size_t)BN * ES);
  w.wbuf = alloc(NP); w.gbuf = alloc((size_t)BN * (XS + XV));

  // masks
  mask_init_kernel<<<gblk(BN), 256, 0, stream>>>(masks, w.mf);
  msum_kernel<<<1, B, 0, stream>>>(w.mf, w.msum);
  pairm_kernel<<<gblk(NP), 256, 0, stream>>>(w.mf, w.pairm);

  // encoders (e_vec chain & encoder coords are dead code -> skipped)
  run_encoder(stream, enc_p, w, patoms, padjs, pcoords, w.xs_p, w.xv_p, w.es_p);
  run_encoder(stream, enc_r, w, ratoms, radjs, rcoords, w.xs_r, w.xv_r, w.es_r);

  // x gate: g = sigmoid([xs_r, xs_p] @ Wg + bg); combine into r buffers
  gemm_wmma_kernel<<<gtile(BN, XS + XV, 1), 128, 0, stream>>>(
      ConcatA{w.xs_r, w.xs_p, XS, XS, XS}, x_gate.W, XS + XV, 0, x_gate.b,
      w.gbuf, XS + XV, 0, (long)BN, XS + XV, 2 * XS, 2, 0, nullptr);
  xgate_sca_kernel<<<gblk((long)BN * XS), 256, 0, stream>>>(w.gbuf, w.xs_r, w.xs_p);
  xgate_vec_kernel<<<gblk((long)BN * XV * 3), 256, 0, stream>>>(w.gbuf, w.xv_r, w.xv_p);

  // e gate, chunked (reuse w.attn as the g buffer; only gs cols 0..63 needed
  // since e_vec is dead). W is (2*ES, ES+EV) -> ldb = 72, Nc = 64.
  {
    const long CH = NP / 8;  // 36864 rows; CH*ES fits exactly in w.attn
    for (int c = 0; c < 8; ++c) {
      long r0 = c * CH;
      gemm_wmma_kernel<<<gtile(CH, ES, 1), 128, 0, stream>>>(
          ConcatA{w.es_r + r0 * ES, w.es_p + r0 * ES, ES, ES, ES}, e_gate.W,
          ES + EV, 0, e_gate.b, w.attn, ES, 0, CH, ES, 2 * ES, 2, 0, nullptr);
      egate_combine_kernel<<<gblk(CH * ES), 256, 0, stream>>>(
          w.attn, w.es_r, w.es_p, r0 * ES, CH * ES);
    }
  }

  // sigma embedding
  sigemb_kernel<<<gblk((long)BN * XS), 256, 0, stream>>>(w.xs_r, sigmas,
                                                         sig_emb.W, sig_emb.b);
  // decoder
  copy_kernel<<<gblk((long)BN * 3), 256, 0, stream>>>(w.coords, tscoords,
                                                      (long)BN * 3);
  run_layer(stream, dec[0], w, w.xs_r, w.xv_r, w.es_r, w.coords, true);
  run_layer(stream, dec[1], w, w.xs_r, w.xv_r, w.es_r, w.coords, true);

  // out = c_skip*tscoords + c_out*denoised
  final_kernel<<<gblk((long)BN * 3), 256, 0, stream>>>(tscoords, w.coords,
                                                       sigmas, (float*)d_out);
}